// GetEdgeJK_80659485818992
// MI455X (gfx1250) — compile-verified
//
#include <hip/hip_runtime.h>
#include <stdint.h>

// Problem constants (from the reference)
#define B_    4
#define AT_   384
#define NBR_  24
#define FEAT_ 128
#define NT_   256   // threads per block (8 waves of 32)

typedef unsigned int u32x4 __attribute__((ext_vector_type(4)));
typedef int          i32x8 __attribute__((ext_vector_type(8)));
typedef int          i32x4 __attribute__((ext_vector_type(4)));
typedef float        f32x4 __attribute__((ext_vector_type(4)));

__global__ __launch_bounds__(NT_) void edge_jk_kernel(
    const float* __restrict__ edge,   // (B, AT, NBR, FEAT) f32
    const int*   __restrict__ nbr,    // (B, AT, NBR) i32
    const float* __restrict__ cell,   // (B, AT, NBR, 3) f32
    float*       __restrict__ out)    // (B, AT, NBR, NBR, FEAT) f32
{
    __shared__ __align__(16) float s_edge[NBR_ * FEAT_]; // 12 KB, filled by TDM
    __shared__ int   s_nbrA[NBR_];            // nbr_idx[b,a,:]
    __shared__ float s_cellA[NBR_ * 3];       // cell_offset[b,a,:,:]
    __shared__ int   s_jnbr[NBR_ * NBR_];     // nbr_idx[b, nbr_idx[b,a,j], k]
    __shared__ float s_cellJ[NBR_ * NBR_ * 3];// cell_offset[b, nbr_idx[b,a,j], k, :]
    __shared__ int   s_mask[NBR_ * NBR_];     // mask[j,i]

    const int ba  = blockIdx.x;     // flat (b*AT + a), 0..1535
    const int tid = threadIdx.x;

    // ---- Phase 0: kick off TDM DMA of edge_embedding[b,a,:,:] (3072 f32, 12KB)
    // into LDS. One wave issues it (TDM ignores EXEC; issue is per-wave).
    if (tid < 32) {
        // Low 32 bits of the flat address of an LDS object are the LDS byte
        // offset (ISA 10.2).
        unsigned lds_addr = (unsigned)(uintptr_t)(&s_edge[0]);
        unsigned long long gaddr =
            (unsigned long long)(uintptr_t)(edge + (size_t)ba * (NBR_ * FEAT_));

        // D# Group 0 (128b): count=1 (valid), lds_addr, 57-bit global addr, type=2
        u32x4 g0;
        g0[0] = 1u;                                   // [1:0] count=1, rest 0
        g0[1] = lds_addr;                             // [63:32] lds_addr
        g0[2] = (unsigned)(gaddr & 0xFFFFFFFFu);      // [95:64] global_addr lo
        g0[3] = ((unsigned)(gaddr >> 32) & 0x01FFFFFFu) | 0x80000000u; // addr hi | type=2

        // D# Group 1 (256b): data_size=4B; 1-D copy of 3072 elements.
        i32x8 g1;
        g1[0] = 0x00020000;            // [17:16] data_size=2 (4B); mask/pad/iter=0
        g1[1] = (NBR_ * FEAT_) << 16;  // [79:48] tensor_dim0 = 3072 (low 16 here)
        g1[2] = 1 << 16;               // [111:80] tensor_dim1 = 1
        g1[3] = (NBR_ * FEAT_) << 16;  // [127:112] tile_dim0 = 3072
        g1[4] = 1;                     // [143:128] tile_dim1 = 1 (single row)
        g1[5] = NBR_ * FEAT_;          // [207:160] tensor_dim0_stride = 3072
        g1[6] = 0;
        g1[7] = 0;

        i32x4 z4 = {0, 0, 0, 0};
#if defined(__clang_major__) && (__clang_major__ >= 23)
        i32x8 z8 = {0, 0, 0, 0, 0, 0, 0, 0};
        __builtin_amdgcn_tensor_load_to_lds(g0, g1, z4, z4, z8, 0);
#else
        __builtin_amdgcn_tensor_load_to_lds(g0, g1, z4, z4, 0);
#endif
    }

    // ---- Phase 1 (overlapped with DMA): stage this atom's index/cell rows.
    for (int v = tid; v < NBR_; v += NT_)      s_nbrA[v]  = nbr[ba * NBR_ + v];
    for (int v = tid; v < NBR_ * 3; v += NT_)  s_cellA[v] = cell[ba * NBR_ * 3 + v];
    __syncthreads();

    const int b = ba / AT_;

    // Gather neighbor-of-neighbor index + cell rows: 576 (j,k) pairs.
    for (int v = tid; v < NBR_ * NBR_; v += NT_) {
        const int j  = v / NBR_;
        const int k  = v - j * NBR_;
        const int j2 = s_nbrA[j];                      // nbr_idx[b,a,j]
        const int base = (b * AT_ + j2) * NBR_ + k;
        s_jnbr[v]        = nbr[base];
        s_cellJ[v * 3 + 0] = cell[base * 3 + 0];
        s_cellJ[v * 3 + 1] = cell[base * 3 + 1];
        s_cellJ[v * 3 + 2] = cell[base * 3 + 2];
    }
    __syncthreads();

    // ---- Phase 2: mask[j,i] = any_k ( j_nbr[j,k]==target(i,j) && cell match )
    for (int v = tid; v < NBR_ * NBR_; v += NT_) {
        const int j = v / NBR_;
        const int i = v - j * NBR_;
        const int target = (i == j) ? -1 : s_nbrA[i];
        const float c0 = s_cellA[i * 3 + 0];
        const float c1 = s_cellA[i * 3 + 1];
        const float c2 = s_cellA[i * 3 + 2];
        int m = 0;
#pragma unroll
        for (int k = 0; k < NBR_; ++k) {
            const int jk = j * NBR_ + k;
            m |= (int)((s_jnbr[jk] == target) &
                       (s_cellJ[jk * 3 + 0] == c0) &
                       (s_cellJ[jk * 3 + 1] == c1) &
                       (s_cellJ[jk * 3 + 2] == c2));
        }
        s_mask[v] = m;
    }

    // DMA must have landed before any wave reads s_edge after the barrier.
    if (tid < 32) __builtin_amdgcn_s_wait_tensorcnt(0);
    __syncthreads();

    // ---- Phase 3: stream 24*24 rows x 128 f32 = 288 KB per tile.
    // 18432 16B stores; 32 consecutive lanes cover one 128-float row
    // -> coalesced global_store_b128, non-temporal (output is write-once).
    const f32x4* s_e4 = (const f32x4*)s_edge;
    f32x4* out4 = (f32x4*)(out + (size_t)ba * (NBR_ * NBR_ * FEAT_));
    const f32x4 zero4 = {0.f, 0.f, 0.f, 0.f};
#pragma unroll 4
    for (int v = tid; v < NBR_ * NBR_ * (FEAT_ / 4); v += NT_) {
        const int row = v >> 5;          // FEAT/4 == 32 vec4 per row
        const int f   = v & 31;
        const int i   = row % NBR_;
        const f32x4 val = s_mask[row] ? s_e4[i * (FEAT_ / 4) + f] : zero4;
        __builtin_nontemporal_store(val, &out4[v]);
    }
}

extern "C" void kernel_launch(void* const* d_in, const int* in_sizes, int n_in,
                              void* d_out, int out_size, void* d_ws, size_t ws_size,
                              hipStream_t stream) {
    const float* edge = (const float*)d_in[0];   // edge_embedding f32
    const int*   nbr  = (const int*)d_in[1];     // nbr_idx (integer -> i32 per harness)
    const float* cell = (const float*)d_in[2];   // cell_offset f32
    float* out = (float*)d_out;

    dim3 grid(B_ * AT_);                          // 1536 tiles
    dim3 block(NT_);
    edge_jk_kernel<<<grid, block, 0, stream>>>(edge, nbr, cell, out);
}